// DiceLoss_38766374814163
// MI455X (gfx1250) — compile-verified
//
#include <hip/hip_runtime.h>
#include <hip/hip_bf16.h>

// Dice loss for [8,16,512,512] f32 logits + [8,512,512] int targets.
// Memory-bound streaming reduction (~136MB -> ~5.9us roofline at 23.3TB/s).
// Pass 1: coalesced float4 loads, register softmax, per-thread P/I/T[16],
//         wave32 shfl_xor butterfly, per-block partials to workspace.
// Pass 2: single wave reduces 1024x48 partials with V_WMMA_F32_16X16X4_F32
//         (A = class x block partials, B = ones) and computes the scalar loss.

#define NUM_CLASSES 16
#define IGNORE_INDEX 255
#define HWS 262144          // 512*512 pixels per image
#define THREADS 256
#define PPB 2048            // pixels per block (256 threads * 8)

typedef __attribute__((ext_vector_type(2))) float v2f;
typedef __attribute__((ext_vector_type(8))) float v8f;

__global__ __launch_bounds__(THREADS)
void dice_partial_kernel(const float* __restrict__ logits,
                         const int*   __restrict__ targets,
                         float*       __restrict__ partials) {
    const int blk  = blockIdx.x;
    const int bpi  = HWS / PPB;              // blocks per batch image (128)
    const int b    = blk / bpi;
    const int pix0 = (blk % bpi) * PPB;
    const float* lbase = logits + (size_t)b * NUM_CLASSES * HWS;
    const int*   tbase = targets + (size_t)b * HWS;

    float P[NUM_CLASSES], I[NUM_CLASSES], T[NUM_CLASSES];
#pragma unroll
    for (int c = 0; c < NUM_CLASSES; ++c) { P[c] = 0.f; I[c] = 0.f; T[c] = 0.f; }

#pragma unroll
    for (int g = 0; g < 2; ++g) {
        // group g: 1024 contiguous pixels, each thread takes 4 consecutive ones
        const int p = pix0 + g * (THREADS * 4) + threadIdx.x * 4;

        // load 16 classes x 4 pixels (global_load_b128, fully coalesced)
        float xr[4][NUM_CLASSES];
#pragma unroll
        for (int c = 0; c < NUM_CLASSES; ++c) {
            float4 v = *(const float4*)(lbase + (size_t)c * HWS + p);
            xr[0][c] = v.x; xr[1][c] = v.y; xr[2][c] = v.z; xr[3][c] = v.w;
        }
        int4 tg = *(const int4*)(tbase + p);
        int tv[4] = {tg.x, tg.y, tg.z, tg.w};

#pragma unroll
        for (int j = 0; j < 4; ++j) {
            float m = xr[j][0];
#pragma unroll
            for (int c = 1; c < NUM_CLASSES; ++c) m = fmaxf(m, xr[j][c]);
            float e[NUM_CLASSES];
            float s = 0.f;
#pragma unroll
            for (int c = 0; c < NUM_CLASSES; ++c) { e[c] = __expf(xr[j][c] - m); s += e[c]; }
            const float r = 1.0f / s;
            const int   t = tv[j];
            // t == c implies valid since IGNORE_INDEX (255) >= NUM_CLASSES
#pragma unroll
            for (int c = 0; c < NUM_CLASSES; ++c) {
                const float prob = e[c] * r;
                const float hit  = (t == c) ? 1.0f : 0.0f;
                P[c] += prob;                 // p_sum (unmasked, as reference)
                I[c]  = fmaf(hit, prob, I[c]);
                T[c] += hit;
            }
        }
    }

    // wave32 butterfly reduction of all 48 accumulators
#pragma unroll
    for (int off = 16; off > 0; off >>= 1) {
#pragma unroll
        for (int c = 0; c < NUM_CLASSES; ++c) {
            P[c] += __shfl_xor(P[c], off, 32);
            I[c] += __shfl_xor(I[c], off, 32);
            T[c] += __shfl_xor(T[c], off, 32);
        }
    }

    __shared__ float smem[THREADS / 32][48];
    const int wave = threadIdx.x >> 5;
    const int lane = threadIdx.x & 31;
    if (lane == 0) {
#pragma unroll
        for (int c = 0; c < NUM_CLASSES; ++c) {
            smem[wave][c]      = P[c];
            smem[wave][16 + c] = I[c];
            smem[wave][32 + c] = T[c];
        }
    }
    __syncthreads();
    if (threadIdx.x < 48) {
        float s = 0.f;
#pragma unroll
        for (int w = 0; w < THREADS / 32; ++w) s += smem[w][threadIdx.x];
        partials[(size_t)blk * 48 + threadIdx.x] = s;
    }
}

// One wave. Reduces nblocks x 48 partials with V_WMMA_F32_16X16X4_F32:
// D[m][n] += sum_k A[m][k] * 1  where A[m][k] = partial(class m, block 4j+k).
__global__ __launch_bounds__(32)
void dice_final_kernel(const float* __restrict__ partials,
                       float*       __restrict__ out,
                       int nblocks) {
    const int lane = threadIdx.x;
    const int l    = lane & 15;           // matrix row M = class (A) / column N (B,D)
    const int k0   = (lane >> 4) * 2;     // lanes 0-15 hold K=0,1; lanes 16-31 hold K=2,3

    v8f accP = {}; v8f accI = {}; v8f accT = {};
    v2f bones; bones.x = 1.0f; bones.y = 1.0f;   // B = ones(4x16)

    for (int j = 0; j < nblocks; j += 4) {
        const float* r0 = partials + (size_t)(j + k0) * 48;
        const float* r1 = partials + (size_t)(j + k0 + 1) * 48;
        v2f aP, aI, aT;
        aP.x = r0[l];      aP.y = r1[l];
        aI.x = r0[16 + l]; aI.y = r1[16 + l];
        aT.x = r0[32 + l]; aT.y = r1[32 + l];
        // 8 args: (neg_a, A, neg_b, B, c_mod, C, reuse_a, reuse_b)
        accP = __builtin_amdgcn_wmma_f32_16x16x4_f32(false, aP, false, bones, (short)0, accP, false, false);
        accI = __builtin_amdgcn_wmma_f32_16x16x4_f32(false, aI, false, bones, (short)0, accI, false, false);
        accT = __builtin_amdgcn_wmma_f32_16x16x4_f32(false, aT, false, bones, (short)0, accT, false, false);
    }

    // D layout: lane n (n<16) VGPR j holds D[M=j][N=n]; lane n+16 holds D[M=8+j][N=n].
    // All columns identical (B = ones), so lane 0 has classes 0-7, lane 16 has 8-15.
    __shared__ float sm[48];
    if (lane == 0) {
#pragma unroll
        for (int j = 0; j < 8; ++j) { sm[j] = accP[j]; sm[16 + j] = accI[j]; sm[32 + j] = accT[j]; }
    }
    if (lane == 16) {
#pragma unroll
        for (int j = 0; j < 8; ++j) { sm[8 + j] = accP[j]; sm[24 + j] = accI[j]; sm[40 + j] = accT[j]; }
    }
    __syncthreads();
    if (lane == 0) {
        float loss = 0.f;
#pragma unroll
        for (int c = 0; c < NUM_CLASSES; ++c) {
            const float Pc = sm[c], Ic = sm[16 + c], Tc = sm[32 + c];
            const float dice = (2.0f * Ic + 1.0f) / (Pc + Tc + 1.0f);
            loss += 1.0f - dice;
        }
        out[0] = loss / (float)NUM_CLASSES;
    }
}

extern "C" void kernel_launch(void* const* d_in, const int* in_sizes, int n_in,
                              void* d_out, int out_size, void* d_ws, size_t ws_size,
                              hipStream_t stream) {
    const float* logits  = (const float*)d_in[0];
    const int*   targets = (const int*)d_in[1];
    float*       out      = (float*)d_out;
    float*       partials = (float*)d_ws;   // nblocks * 48 floats (192 KB for 1024 blocks)

    const int npix    = in_sizes[1];        // B*H*W = 2,097,152
    const int nblocks = npix / PPB;         // 1024 (multiple of 4)

    dice_partial_kernel<<<nblocks, THREADS, 0, stream>>>(logits, targets, partials);
    dice_final_kernel<<<1, 32, 0, stream>>>(partials, out, nblocks);
}